// M_72842645340510
// MI455X (gfx1250) — compile-verified
//
#include <hip/hip_runtime.h>
#include <hip/hip_bf16.h>
#include <math.h>

typedef __attribute__((ext_vector_type(2))) float v2f;
typedef __attribute__((ext_vector_type(8))) float v8f;

#define CDIM 336
#define CR   84
#define BDIM 256
#define HWSZ 196      // 14*14
#define EPSV 1e-5f

#define MB   48       // output channels per workgroup in GEMM (3 M-tiles)
#define NMB  7        // 336 / 48
#define AST  340      // padded LDS row stride (conflict-free for 16-lane groups)
#define NTILES 13     // ceil(196/16)

// ---------------- Kernel 1: SE gate --------------------------------------
__global__ __launch_bounds__(128) void se_gate_kernel(
    const float* __restrict__ x160, const float* __restrict__ w51,
    const float* __restrict__ b51,  const float* __restrict__ w52,
    const float* __restrict__ b52,  float* __restrict__ gate) {
  __shared__ float s_s[CDIM];
  __shared__ float s_h[CR];
  const int b = blockIdx.x;
  const int t = threadIdx.x;
  for (int c = t; c < CDIM; c += 128) s_s[c] = x160[b * CDIM + c];
  __syncthreads();
  if (t < CR) {
    float acc = b51[t];
    const float* wr = w51 + t * CDIM;
    for (int c = 0; c < CDIM; ++c) acc = fmaf(s_s[c], wr[c], acc);
    s_h[t] = fmaxf(acc, 0.0f);                       // relu
  }
  __syncthreads();
  for (int c = t; c < CDIM; c += 128) {
    float acc = b52[c];
    const float* wr = w52 + c * CR;
    for (int j = 0; j < CR; ++j) acc = fmaf(s_h[j], wr[j], acc);
    gate[b * CDIM + c] = 1.0f / (1.0f + __expf(-acc)); // sigmoid
  }
}

// ---------------- Kernel 2: fused gate*conv1x1 GEMM via f32 WMMA ---------
// x166[b,o,p] = sum_c (w53[o,c]*gate[b,c]) * x159[b,c,p]
// Each wave owns a 48(M) x 16(N) strip: one B fragment feeds 3 WMMAs.
__global__ __launch_bounds__(256) void gemm_wmma_kernel(
    const float* __restrict__ x159, const float* __restrict__ w53,
    const float* __restrict__ gate, float* __restrict__ x166) {
  __shared__ float sA[MB * AST];
  const int mblk = blockIdx.x;           // 0..6
  const int b    = blockIdx.y;           // 0..255
  const int t    = threadIdx.x;
  const int m0   = mblk * MB;

  // Stage A' = gate (row-broadcast) * w53 block into LDS
  for (int idx = t; idx < MB * CDIM; idx += 256) {
    const int i = idx / CDIM;
    const int c = idx - i * CDIM;
    sA[i * AST + c] = w53[(m0 + i) * CDIM + c] * gate[b * CDIM + c];
  }
  __syncthreads();

  const int wave = t >> 5;
  const int lane = t & 31;
  const int half = lane >> 4;            // 0: lanes 0-15, 1: lanes 16-31
  const int lid  = lane & 15;
  const float* xb = x159 + (size_t)b * CDIM * HWSZ;

  for (int nt = wave; nt < NTILES; nt += 8) {             // wave-uniform
    const int n0 = nt * 16;
    const int coln = n0 + lid;
    const int colc = (coln < HWSZ) ? coln : (HWSZ - 1);   // clamp OOB cols (masked at store)
    const float* bp = xb + colc;
    const float* ap = &sA[lid * AST];                     // A row = lane%16 (M-tile 0)

    __builtin_prefetch(bp, 0, 0);                          // global_prefetch_b8
    __builtin_prefetch(bp + (size_t)168 * HWSZ, 0, 0);

    v8f acc0 = {0.f, 0.f, 0.f, 0.f, 0.f, 0.f, 0.f, 0.f};
    v8f acc1 = acc0;
    v8f acc2 = acc0;
    #pragma unroll 4
    for (int k = 0; k < CDIM; k += 4) {
      const int kb = k + half * 2;                         // ISA A layout: V0=K0/K2, V1=K1/K3
      v2f bb;
      bb.x = bp[(size_t)kb * HWSZ];                        // B row K=kb,   col = lane%16
      bb.y = bp[(size_t)(kb + 1) * HWSZ];                  // B row K=kb+1
      v2f a0 = *(const v2f*)(ap + kb);                     // ds_load_b64 (pairs to 2addr)
      v2f a1 = *(const v2f*)(ap + 16 * AST + kb);
      v2f a2 = *(const v2f*)(ap + 32 * AST + kb);
      acc0 = __builtin_amdgcn_wmma_f32_16x16x4_f32(false, a0, false, bb, (short)0, acc0, false, false);
      acc1 = __builtin_amdgcn_wmma_f32_16x16x4_f32(false, a1, false, bb, (short)0, acc1, false, false);
      acc2 = __builtin_amdgcn_wmma_f32_16x16x4_f32(false, a2, false, bb, (short)0, acc2, false, false);
    }

    // C/D layout: VGPR v -> row v + 8*half, col = lane%16
    if (coln < HWSZ) {
      float* op = x166 + ((size_t)b * CDIM + m0) * HWSZ + n0 + lid;
      #pragma unroll
      for (int v = 0; v < 8; ++v) {
        const int row = v + half * 8;
        op[(size_t)(row +  0) * HWSZ] = acc0[v];
        op[(size_t)(row + 16) * HWSZ] = acc1[v];
        op[(size_t)(row + 32) * HWSZ] = acc2[v];
      }
    }
  }
}

// ---------------- Kernel 3: per-channel mean/var -> scale/shift ----------
__global__ __launch_bounds__(256) void stats_kernel(
    const float* __restrict__ x166, const float* __restrict__ gamma,
    const float* __restrict__ beta, float* __restrict__ stats) {
  __shared__ float s_sum[256];
  __shared__ float s_sq[256];
  const int o = blockIdx.x;
  const int t = threadIdx.x;
  float sum = 0.f, sq = 0.f;
  for (int b = 0; b < BDIM; ++b) {
    const float* p = x166 + ((size_t)b * CDIM + o) * HWSZ;
    for (int i = t; i < HWSZ; i += 256) {
      const float v = p[i];
      sum += v;
      sq = fmaf(v, v, sq);
    }
  }
  s_sum[t] = sum; s_sq[t] = sq;
  __syncthreads();
  for (int st = 128; st > 0; st >>= 1) {     // fixed-order tree: deterministic
    if (t < st) { s_sum[t] += s_sum[t + st]; s_sq[t] += s_sq[t + st]; }
    __syncthreads();
  }
  if (t == 0) {
    const float inv_n = 1.0f / (float)(BDIM * HWSZ);
    const float mean = s_sum[0] * inv_n;
    const float var  = s_sq[0] * inv_n - mean * mean;
    const float sc   = gamma[o] * rsqrtf(var + EPSV);
    stats[2 * o]     = sc;
    stats[2 * o + 1] = beta[o] - mean * sc;
  }
}

// ---------------- Kernel 4: in-place normalize (float4) ------------------
__global__ __launch_bounds__(256) void norm_kernel(
    float* __restrict__ out, const float* __restrict__ stats) {
  const size_t N4 = (size_t)BDIM * CDIM * HWSZ / 4;
  const size_t idx4 = (size_t)blockIdx.x * 256 + threadIdx.x;
  if (idx4 >= N4) return;
  const size_t e = idx4 * 4;                         // rows are 49 float4s: never split
  const int o = (int)((e / HWSZ) % CDIM);
  const float sc = stats[2 * o];
  const float sh = stats[2 * o + 1];
  float4 v = *(const float4*)(out + e);
  v.x = fmaf(v.x, sc, sh);
  v.y = fmaf(v.y, sc, sh);
  v.z = fmaf(v.z, sc, sh);
  v.w = fmaf(v.w, sc, sh);
  *(float4*)(out + e) = v;
}

// ---------------- Host launcher ------------------------------------------
extern "C" void kernel_launch(void* const* d_in, const int* in_sizes, int n_in,
                              void* d_out, int out_size, void* d_ws, size_t ws_size,
                              hipStream_t stream) {
  const float* x160  = (const float*)d_in[0];
  const float* x159  = (const float*)d_in[1];
  const float* w51   = (const float*)d_in[2];
  const float* b51   = (const float*)d_in[3];
  const float* w52   = (const float*)d_in[4];
  const float* b52   = (const float*)d_in[5];
  const float* w53   = (const float*)d_in[6];
  const float* gamma = (const float*)d_in[7];
  const float* beta  = (const float*)d_in[8];
  float* out = (float*)d_out;

  float* gate  = (float*)d_ws;                 // B*C floats
  float* stats = gate + (size_t)BDIM * CDIM;   // C*2 floats

  se_gate_kernel<<<BDIM, 128, 0, stream>>>(x160, w51, b51, w52, b52, gate);

  dim3 g2(NMB, BDIM);
  gemm_wmma_kernel<<<g2, 256, 0, stream>>>(x159, w53, gate, out);

  stats_kernel<<<CDIM, 256, 0, stream>>>(out, gamma, beta, stats);

  const size_t N4 = (size_t)BDIM * CDIM * HWSZ / 4;
  norm_kernel<<<(int)((N4 + 255) / 256), 256, 0, stream>>>(out, stats);
}